// mysoftmax_49220325212429
// MI455X (gfx1250) — compile-verified
//
#include <hip/hip_runtime.h>

// Softmax + variance propagation through the softmax Jacobian.
//   p = softmax(mu)                      (per pixel, C = 19)
//   sigma_out_i = p_i^2 * ((1 - 2 p_i) * sigma_i + S),  S = sum_j p_j^2 sigma_j
// Memory-bound: ~80 MB total traffic -> ~3.4 us at 23.3 TB/s. Strategy:
// cooperative async B128 global->LDS staging (CDNA5 ASYNCcnt path),
// conflict-free LDS compute (stride 19 is coprime with 64 banks),
// async B128 LDS->global writeback.

#define CCH 19
#define PIX 256
#define TILE (CCH * PIX)   // 4864 floats = 1216 x 16B chunks, tile base 16B-aligned

#if defined(__has_builtin)
#  if __has_builtin(__builtin_amdgcn_global_load_async_to_lds_b128) && \
      __has_builtin(__builtin_amdgcn_global_load_async_to_lds_b32)  && \
      __has_builtin(__builtin_amdgcn_global_store_async_from_lds_b128) && \
      __has_builtin(__builtin_amdgcn_global_store_async_from_lds_b32)
#    define USE_ASYNC_LDS 1
#  endif
#endif
#ifndef USE_ASYNC_LDS
#  define USE_ASYNC_LDS 0
#endif

// Param types per hipcc diagnostics: arg0 = global (AS1) int4* / int*,
// arg1 = LDS (AS3) pointer, args 2/3 = immediate offset / cpol.
typedef int v4i __attribute__((vector_size(16)));
typedef __attribute__((address_space(1))) v4i* gv4i_p;
typedef __attribute__((address_space(3))) v4i* lv4i_p;
typedef __attribute__((address_space(1))) int* gi_p;
typedef __attribute__((address_space(3))) int* li_p;

__device__ __forceinline__ void wait_asynccnt0() {
#if defined(__has_builtin) && __has_builtin(__builtin_amdgcn_s_wait_asynccnt)
  __builtin_amdgcn_s_wait_asynccnt(0);
#else
  asm volatile("s_wait_asynccnt 0" ::: "memory");
#endif
}

extern "C" __global__ void __launch_bounds__(PIX)
softmax_jacprop_kernel(const float* __restrict__ mu_g,
                       const float* __restrict__ sig_g,
                       float* __restrict__ p_g,
                       float* __restrict__ so_g,
                       int npix)
{
  __shared__ __align__(16) float smu[TILE];
  __shared__ __align__(16) float ssg[TILE];

  const int tid = threadIdx.x;
  const int pixBase = blockIdx.x * PIX;
  int validPix = npix - pixBase;
  if (validPix <= 0) return;
  if (validPix > PIX) validPix = PIX;

  const long long fBase = (long long)pixBase * CCH;
  const int validF = validPix * CCH;   // == TILE for full blocks
  const int nv4 = validF >> 2;         // full 16B chunks

  const float* gm = mu_g  + fBase;
  const float* gs = sig_g + fBase;

  // ---------------- stage inputs into LDS ----------------
#if USE_ASYNC_LDS
  for (int v = tid; v < nv4; v += PIX) {
    const int f = v << 2;
    __builtin_amdgcn_global_load_async_to_lds_b128(
        (gv4i_p)(gm + f), (lv4i_p)(smu + f), 0, 0);
    __builtin_amdgcn_global_load_async_to_lds_b128(
        (gv4i_p)(gs + f), (lv4i_p)(ssg + f), 0, 0);
  }
  for (int f = (nv4 << 2) + tid; f < validF; f += PIX) {  // scalar tail (partial blocks only)
    __builtin_amdgcn_global_load_async_to_lds_b32(
        (gi_p)(gm + f), (li_p)(smu + f), 0, 0);
    __builtin_amdgcn_global_load_async_to_lds_b32(
        (gi_p)(gs + f), (li_p)(ssg + f), 0, 0);
  }
  wait_asynccnt0();
#else
  {
    const float4* gm4 = reinterpret_cast<const float4*>(gm);
    const float4* gs4 = reinterpret_cast<const float4*>(gs);
    for (int v = tid; v < nv4; v += PIX) {
      float4 a = gm4[v];
      float4 b = gs4[v];
      *reinterpret_cast<float4*>(smu + (v << 2)) = a;
      *reinterpret_cast<float4*>(ssg + (v << 2)) = b;
    }
    for (int f = (nv4 << 2) + tid; f < validF; f += PIX) {
      smu[f] = gm[f];
      ssg[f] = gs[f];
    }
  }
#endif
  __syncthreads();

  // ---------------- per-pixel compute (one thread per pixel) ----------------
  if (tid < validPix) {
    const int base = tid * CCH;         // stride 19: conflict-free on 64 banks
    float x[CCH];
#pragma unroll
    for (int c = 0; c < CCH; ++c) x[c] = smu[base + c];

    float m = x[0];
#pragma unroll
    for (int c = 1; c < CCH; ++c) m = fmaxf(m, x[c]);

    float den = 0.0f;
#pragma unroll
    for (int c = 0; c < CCH; ++c) { x[c] = __expf(x[c] - m); den += x[c]; }
    const float r = 1.0f / den;

    float s[CCH];
#pragma unroll
    for (int c = 0; c < CCH; ++c) s[c] = ssg[base + c];

    float S = 0.0f;
#pragma unroll
    for (int c = 0; c < CCH; ++c) {
      x[c] *= r;                         // x[c] = p_c
      S = fmaf(x[c] * x[c], s[c], S);    // S += p^2 * sigma
    }
#pragma unroll
    for (int c = 0; c < CCH; ++c) {
      const float p  = x[c];
      const float pp = p * p;
      const float so = pp * fmaf(1.0f - 2.0f * p, s[c], S);
      smu[base + c] = p;                 // private region: safe in-place reuse
      ssg[base + c] = so;
    }
  }
  __syncthreads();

  // ---------------- write outputs from LDS ----------------
  float* gp = p_g  + fBase;
  float* go = so_g + fBase;
#if USE_ASYNC_LDS
  for (int v = tid; v < nv4; v += PIX) {
    const int f = v << 2;
    __builtin_amdgcn_global_store_async_from_lds_b128(
        (gv4i_p)(gp + f), (lv4i_p)(smu + f), 0, 0);
    __builtin_amdgcn_global_store_async_from_lds_b128(
        (gv4i_p)(go + f), (lv4i_p)(ssg + f), 0, 0);
  }
  for (int f = (nv4 << 2) + tid; f < validF; f += PIX) {
    __builtin_amdgcn_global_store_async_from_lds_b32(
        (gi_p)(gp + f), (li_p)(smu + f), 0, 0);
    __builtin_amdgcn_global_store_async_from_lds_b32(
        (gi_p)(go + f), (li_p)(ssg + f), 0, 0);
  }
  wait_asynccnt0();
#else
  {
    float4* gp4 = reinterpret_cast<float4*>(gp);
    float4* go4 = reinterpret_cast<float4*>(go);
    for (int v = tid; v < nv4; v += PIX) {
      gp4[v] = *reinterpret_cast<const float4*>(smu + (v << 2));
      go4[v] = *reinterpret_cast<const float4*>(ssg + (v << 2));
    }
    for (int f = (nv4 << 2) + tid; f < validF; f += PIX) {
      gp[f] = smu[f];
      go[f] = ssg[f];
    }
  }
#endif
}

extern "C" void kernel_launch(void* const* d_in, const int* in_sizes, int n_in,
                              void* d_out, int out_size, void* d_ws, size_t ws_size,
                              hipStream_t stream) {
  (void)n_in; (void)out_size; (void)d_ws; (void)ws_size;
  const float* mu  = (const float*)d_in[0];
  const float* sig = (const float*)d_in[1];
  float* out = (float*)d_out;

  const long long total = (long long)in_sizes[0];      // B*H*W*C
  const int npix = (int)(total / CCH);                 // 262144

  float* p_out  = out;                                 // first output, flat
  float* so_out = out + (size_t)npix * CCH;            // second output, flat

  const int grid = (npix + PIX - 1) / PIX;             // 1024 full blocks
  softmax_jacprop_kernel<<<grid, PIX, 0, stream>>>(mu, sig, p_out, so_out, npix);
}